// HiddenMarkovStateInferenceLayer_26362509263161
// MI455X (gfx1250) — compile-verified
//
#include <hip/hip_runtime.h>

// ---------------------------------------------------------------------------
// HMM forward-backward (Baum-Welch posteriors) for gfx1250 / MI455X.
//
// Prob-domain scaled recursions (per-step scalar factors cancel in gamma/xi):
//   alpha_t = rescale( (alpha_{t-1} P) .* E_t )          E = exp(log_B)
//   beta_t  = rescale(  P (E_{t+1} .* beta_{t+1}) )
//   gamma_t = normalize( alpha_t .* beta_t )
//   xi_t    = normalize_jk( alpha_t[j] P[j,k] E_{t+1}[k] beta_{t+1}[k] )
//
// Rescaling only every 4 steps (range control only); exact normalization is
// done once, per (b,t), in the bandwidth-bound posterior phase.
// The 16-state matvecs for 16 batched chains per wave are done as a
// 16x16x16 f32 matmul = 4 chained V_WMMA_F32_16X16X4_F32.
// ---------------------------------------------------------------------------

typedef __attribute__((ext_vector_type(2))) float v2f;
typedef __attribute__((ext_vector_type(4))) float v4f;
typedef __attribute__((ext_vector_type(8))) float v8f;

constexpr int NS = 16;    // states
constexpr int TT = 2000;  // sequence length
constexpr int NB = 128;   // batch

__device__ __forceinline__ float fast_rcp(float x) {
  return __builtin_amdgcn_rcpf(x);
}

// 16x16x16 f32 matmul: C[m,n] = sum_j A[m,j] * B[j,n]
// A preloaded in A-layout chunks (lane l: m=l&15, kk=2*(l>>4)+v), B given in
// "S-layout": S[2q+v] holds B[4q + 2*(l>>4) + v, l&15].
__device__ __forceinline__ v8f mm16x16(const v2f A[4], const float S[8]) {
  v8f c = {};
#pragma unroll
  for (int q = 0; q < 4; ++q) {
    v2f b;
    b[0] = S[2 * q + 0];
    b[1] = S[2 * q + 1];
    // (neg_a, A, neg_b, B, c_mod, C, reuse_a, reuse_b)
    c = __builtin_amdgcn_wmma_f32_16x16x4_f32(false, A[q], false, b,
                                              (short)0, c, false, false);
  }
  return c;
}

// Convert a 16x16 result from C-layout (VGPR r, lane l -> row r+8*(l>>4),
// col l&15) into S-layout (B operand for the next WMMA).
// Pairs (c0,c2),(c1,c3),(c4,c6),(c5,c7) feed complementary lane halves, so a
// pre-select lets ONE xor-16 swap serve two S outputs: 4 cross-lane ops total.
__device__ __forceinline__ void c_to_s(const float c[8], float S[8], int h) {
  float z0 = h ? c[0] : c[2];
  float z1 = h ? c[1] : c[3];
  float z2 = h ? c[4] : c[6];
  float z3 = h ? c[5] : c[7];
  float w0 = __shfl_xor(z0, 16, 32);
  float w1 = __shfl_xor(z1, 16, 32);
  float w2 = __shfl_xor(z2, 16, 32);
  float w3 = __shfl_xor(z3, 16, 32);
  S[0] = h ? w0 : c[0];
  S[1] = h ? w1 : c[1];
  S[2] = h ? w2 : c[4];
  S[3] = h ? w3 : c[5];
  S[4] = h ? c[2] : w0;
  S[5] = h ? c[3] : w1;
  S[6] = h ? c[6] : w2;
  S[7] = h ? c[7] : w3;
}

// Rescale the 16 states of each batch column by 1/sum (range control only).
__device__ __forceinline__ void rescale8(float c[8]) {
  float part = ((c[0] + c[1]) + (c[2] + c[3])) + ((c[4] + c[5]) + (c[6] + c[7]));
  float tot = part + __shfl_xor(part, 16, 32);
  float inv = fast_rcp(tot);
#pragma unroll
  for (int r = 0; r < 8; ++r) c[r] *= inv;
}

__device__ __forceinline__ void store8(float* __restrict__ p, const float c[8]) {
  v4f a, b;
#pragma unroll
  for (int i = 0; i < 4; ++i) { a[i] = c[i]; b[i] = c[4 + i]; }
  *(v4f*)p = a;
  *(v4f*)(p + 4) = b;
}

// -------------------- phase 1: E = exp(log_B) ------------------------------
__global__ __launch_bounds__(256) void hmm_exp_kernel(
    const float* __restrict__ in, float* __restrict__ out, int n4) {
  int i = blockIdx.x * blockDim.x + threadIdx.x;
  if (i < n4) {
    v4f x = ((const v4f*)in)[i];
    v4f y;
#pragma unroll
    for (int k = 0; k < 4; ++k) y[k] = __expf(x[k]);
    ((v4f*)out)[i] = y;
  }
}

// -------------------- phase 2: fwd + bwd scans (concurrent) ----------------
// grid = (NB/16, 2), block = 32 (one wave). y==0: forward, y==1: backward.
__global__ __launch_bounds__(32) void hmm_scan_kernel(
    const float* __restrict__ trans, const float* __restrict__ E,
    float* __restrict__ alpha, float* __restrict__ beta) {
  const int lane = threadIdx.x & 31;
  const int h = lane >> 4;
  const int bl = lane & 15;
  const int batch = blockIdx.x * 16 + bl;
  const bool fwd = (blockIdx.y == 0);

  // A operand: forward uses P^T (A[m=k',j]=P[j,k']), backward uses P.
  v2f A[4];
#pragma unroll
  for (int q = 0; q < 4; ++q) {
#pragma unroll
    for (int v = 0; v < 2; ++v) {
      const int kk = 4 * q + 2 * h + v;
      A[q][v] = fwd ? trans[kk * NS + bl] : trans[bl * NS + kk];
    }
  }

  const float* laneE = E + ((size_t)batch * TT) * NS + 8 * h;
  float* laneA = alpha + ((size_t)batch * TT) * NS + 8 * h;
  float* laneB = beta + ((size_t)batch * TT) * NS + 8 * h;

  float S[8];

  if (fwd) {
    // t = 0: alpha_0 = normalize(E_0)   (uniform prior cancels)
    v4f e0 = *(const v4f*)(laneE);
    v4f e1 = *(const v4f*)(laneE + 4);
    float c[8] = {e0[0], e0[1], e0[2], e0[3], e1[0], e1[1], e1[2], e1[3]};
    rescale8(c);
    store8(laneA, c);
    c_to_s(c, S, h);

    const float* ep = laneE + NS;
    float* ap = laneA + NS;
    for (int t = 1; t < TT; ++t, ep += NS, ap += NS) {
      __builtin_prefetch(ep + 8 * NS, 0, 1);
      v4f f0 = *(const v4f*)(ep);
      v4f f1 = *(const v4f*)(ep + 4);
      v8f acc = mm16x16(A, S);  // (alpha P) for 16 chains
      float cc[8];
#pragma unroll
      for (int i = 0; i < 4; ++i) {
        cc[i] = acc[i] * f0[i];
        cc[4 + i] = acc[4 + i] * f1[i];
      }
      if ((t & 3) == 0) rescale8(cc);  // range control only
      store8(ap, cc);
      if (t < TT - 1) c_to_s(cc, S, h);
    }
  } else {
    // t = TT-1: beta = 1/16 everywhere; W = E_{T-1} .* beta
    const float* ep = laneE + (size_t)(TT - 1) * NS;
    v4f e0 = *(const v4f*)(ep);
    v4f e1 = *(const v4f*)(ep + 4);
    float c[8] = {0.0625f, 0.0625f, 0.0625f, 0.0625f,
                  0.0625f, 0.0625f, 0.0625f, 0.0625f};
    store8(laneB + (size_t)(TT - 1) * NS, c);
    float w[8];
#pragma unroll
    for (int i = 0; i < 4; ++i) {
      w[i] = e0[i] * 0.0625f;
      w[4 + i] = e1[i] * 0.0625f;
    }
    c_to_s(w, S, h);

    const float* ept = laneE + (size_t)(TT - 2) * NS;
    float* bp = laneB + (size_t)(TT - 2) * NS;
    for (int t = TT - 2; t >= 0; --t, ept -= NS, bp -= NS) {
      __builtin_prefetch(ept - 8 * NS >= laneE ? ept - 8 * NS : laneE, 0, 1);
      v4f f0 = *(const v4f*)(ept);
      v4f f1 = *(const v4f*)(ept + 4);
      v8f acc = mm16x16(A, S);  // P * (E .* beta)
      float cc[8];
#pragma unroll
      for (int r = 0; r < 8; ++r) cc[r] = acc[r];
      if ((t & 3) == 0) rescale8(cc);  // range control only
      store8(bp, cc);
      if (t > 0) {
#pragma unroll
        for (int i = 0; i < 4; ++i) {
          w[i] = cc[i] * f0[i];
          w[4 + i] = cc[4 + i] * f1[i];
        }
        c_to_s(w, S, h);
      }
    }
  }
}

// -------------------- phase 3: posteriors (bandwidth bound) ----------------
// One wave per (b,t). Lane l: row j=l>>1, cols k0..k0+7 with k0=(l&1)*8.
__global__ __launch_bounds__(256) void hmm_posterior_kernel(
    const float* __restrict__ trans, const float* __restrict__ E,
    const float* __restrict__ alpha, const float* __restrict__ beta,
    float* __restrict__ gamma_out, float* __restrict__ xi_out) {
  const int lane = threadIdx.x & 31;
  const int wave = blockIdx.x * (blockDim.x >> 5) + (threadIdx.x >> 5);
  if (wave >= NB * TT) return;
  const int b = wave / TT;
  const int t = wave - b * TT;
  const size_t base = ((size_t)b * TT + t) * NS;

  // gamma[b,t,:] = normalize(alpha .* beta)
  {
    const int k = lane & 15;
    float g = alpha[base + k] * beta[base + k];
    float s = g;
    s += __shfl_xor(s, 1, 32);
    s += __shfl_xor(s, 2, 32);
    s += __shfl_xor(s, 4, 32);
    s += __shfl_xor(s, 8, 32);
    if (lane < 16) gamma_out[base + k] = g * fast_rcp(s);
  }

  if (t < TT - 1) {
    const int j = lane >> 1;
    const int k0 = (lane & 1) * 8;
    const float aj = alpha[base + j];
    const float* bp = beta + base + NS + k0;
    const float* ep = E + base + NS + k0;
    const float* pp = trans + j * NS + k0;
    v4f b0 = *(const v4f*)bp, b1 = *(const v4f*)(bp + 4);
    v4f e0 = *(const v4f*)ep, e1 = *(const v4f*)(ep + 4);
    v4f p0 = *(const v4f*)pp, p1 = *(const v4f*)(pp + 4);
    float m[8];
#pragma unroll
    for (int i = 0; i < 4; ++i) {
      m[i] = aj * p0[i] * (e0[i] * b0[i]);
      m[4 + i] = aj * p1[i] * (e1[i] * b1[i]);
    }
    float z = ((m[0] + m[1]) + (m[2] + m[3])) + ((m[4] + m[5]) + (m[6] + m[7]));
    z += __shfl_xor(z, 1, 32);
    z += __shfl_xor(z, 2, 32);
    z += __shfl_xor(z, 4, 32);
    z += __shfl_xor(z, 8, 32);
    z += __shfl_xor(z, 16, 32);
    const float inv = fast_rcp(z);
    const size_t xb = ((size_t)b * (TT - 1) + t) * 256 + (size_t)j * NS + k0;
    v4f o0, o1;
#pragma unroll
    for (int i = 0; i < 4; ++i) {
      o0[i] = m[i] * inv;
      o1[i] = m[4 + i] * inv;
    }
    *(v4f*)(xi_out + xb) = o0;
    *(v4f*)(xi_out + xb + 4) = o1;
  }
}

// ---------------------------------------------------------------------------
extern "C" void kernel_launch(void* const* d_in, const int* in_sizes, int n_in,
                              void* d_out, int out_size, void* d_ws,
                              size_t ws_size, hipStream_t stream) {
  (void)in_sizes; (void)n_in; (void)out_size; (void)ws_size;
  const float* logB = (const float*)d_in[0];   // [128,2000,16] f32
  const float* trans = (const float*)d_in[1];  // [16,16] f32

  const size_t NE = (size_t)NB * TT * NS;  // 4,096,000
  float* ws = (float*)d_ws;
  float* E = ws;                 // exp(log_B)
  float* alpha = ws + NE;        // forward probs (scaled)
  float* beta = ws + 2 * NE;     // backward probs (scaled)

  float* gamma_out = (float*)d_out;
  float* xi_out = gamma_out + NE;  // xi follows gamma in flat output

  const int n4 = (int)(NE / 4);
  hmm_exp_kernel<<<(n4 + 255) / 256, 256, 0, stream>>>(logB, E, n4);

  dim3 grid(NB / 16, 2);  // 8 forward waves + 8 backward waves, concurrent
  hmm_scan_kernel<<<grid, 32, 0, stream>>>(trans, E, alpha, beta);

  const int waves = NB * TT;
  hmm_posterior_kernel<<<(waves + 7) / 8, 256, 0, stream>>>(
      trans, E, alpha, beta, gamma_out, xi_out);
}